// TensorFusion_22119081575288
// MI455X (gfx1250) — compile-verified
//
#include <hip/hip_runtime.h>

typedef float v2f __attribute__((ext_vector_type(2)));
typedef float v8f __attribute__((ext_vector_type(8)));

#define B_ROWS 4096
#define H_DIM  1024
#define OUT_D  21
#define BM     32        // block rows (batch)
#define BN     128       // block cols (hidden)
#define KC     32        // j-chunk per LDS stage
#define ASTRIDE 260      // padded audio-tile row stride (bank-conflict-free)
#define PSTR   288       // dwords per k-pair row of paired W chunk (288%64=32)

// ---------------------------------------------------------------------------
// Kernel 1: h = relu( einsum('bi,bj,ijh->bh', v, a, W1) + b1 )
// GEMM C[4096,1024] = F[4096,65536] @ W1, F built on the fly from v,a.
// 128 threads = 4 waves; wave tile 16(M) x 64(N) -> 4 N-frags per A-build.
// W chunk staged K-pair-interleaved: w_p[k/2][n][k&1] so each B-frag is one
// aligned ds_load_b64 (no register shuffles).
// ---------------------------------------------------------------------------
__global__ __launch_bounds__(128, 1) void fused_bilinear_kernel(
    const float* __restrict__ visual, const float* __restrict__ audio,
    const float* __restrict__ W1, const float* __restrict__ bias1,
    float* __restrict__ hout)
{
  __shared__ float a_s[BM][ASTRIDE];     // 32*260*4 = 33,280 B
  __shared__ float w_p[16 * PSTR];       // 16*288*4 = 18,432 B (paired W chunk)

  const int tid  = threadIdx.x;
  const int wave = tid >> 5;
  const int lane = tid & 31;
  const int ml   = lane & 15;            // position within 16-lane half
  const int hf   = lane >> 4;            // lane half: K-pair select / C M-half
  const int wm   = wave & 1;             // wave row half (16 rows)
  const int wn   = wave >> 1;            // wave col half (64 cols)

  const int row0 = blockIdx.x * BM;      // x walks batch tiles (L2-friendly)
  const int col0 = blockIdx.y * BN;      // y walks hidden tiles

  // ---- stage audio tile once: a_s[r][j] = a(row0+r, j), ones at j=255 ----
  for (int f = tid; f < BM * 256; f += 128) {
    const int r = f >> 8, c = f & 255;
    a_s[r][c] = (c < 255) ? audio[(row0 + r) * 255 + c] : 1.0f;
  }

  // ---- prefetch first W1 chunk (i=0, j=0..31) into registers ----
  float4 wreg[8];
#pragma unroll
  for (int q = 0; q < 8; ++q) {
    const int f = q * 128 + tid;         // 1024 float4s = 32 rows x 128 cols
    const int jj = f >> 5, cc = f & 31;
    wreg[q] = *(const float4*)&W1[(size_t)jj * H_DIM + col0 + cc * 4];
  }

  v8f acc[4];
#pragma unroll
  for (int nf = 0; nf < 4; ++nf)
    acc[nf] = (v8f){0.f, 0.f, 0.f, 0.f, 0.f, 0.f, 0.f, 0.f};

  const int rowi = row0 + wm * 16 + ml;  // this lane's A-matrix row (batch)
  float vv = 1.0f;

  // 2048 chunks: c -> i = c>>3 (visual idx), jbase = (c&7)*32 (audio idx base)
  for (int c = 0; c < 2048; ++c) {
    const int i = c >> 3;
    const int jbase = (c & 7) << 5;

    __syncthreads();                     // previous compute done -> w_p free
#pragma unroll
    for (int q = 0; q < 8; ++q) {        // scatter into paired layout
      const int f = q * 128 + tid;
      const int jj = f >> 5, cc = f & 31;
      const int base = (jj >> 1) * PSTR + cc * 8 + (jj & 1);
      w_p[base + 0] = wreg[q].x;         // merges into ds_store_2addr_b32 x2
      w_p[base + 2] = wreg[q].y;
      w_p[base + 4] = wreg[q].z;
      w_p[base + 6] = wreg[q].w;
    }
    __syncthreads();                     // w_p ready

    if (c + 1 < 2048) {                  // prefetch next chunk during compute
      const int ni = (c + 1) >> 3;
      const int njb = ((c + 1) & 7) << 5;
#pragma unroll
      for (int q = 0; q < 8; ++q) {
        const int f = q * 128 + tid;
        const int jj = f >> 5, cc = f & 31;
        wreg[q] = *(const float4*)&W1[((size_t)ni * 256 + njb + jj) * H_DIM + col0 + cc * 4];
      }
    }

    if ((c & 7) == 0)                    // new i: fetch v(b,i) (uniform branch)
      vv = (i < 255) ? visual[rowi * 255 + i] : 1.0f;

    const int nbase = (wn * 64 + ml) * 2;     // dword offset of this lane's N
#pragma unroll
    for (int kk = 0; kk < KC; kk += 4) {
      // A frag 16x4 f32: lanes 0-15 hold K={0,1}, lanes 16-31 hold K={2,3}
      const float* ap = &a_s[wm * 16 + ml][jbase + kk + 2 * hf];
      v2f af; af.x = vv * ap[0]; af.y = vv * ap[1];     // v_pk_mul_f32
      // B frags: one b64 each from paired layout, conflict-free
      const float* wrow = &w_p[(kk / 2 + hf) * PSTR];
      const v2f b0 = *(const v2f*)&wrow[nbase];
      const v2f b1 = *(const v2f*)&wrow[nbase + 32];
      const v2f b2 = *(const v2f*)&wrow[nbase + 64];
      const v2f b3 = *(const v2f*)&wrow[nbase + 96];
      acc[0] = __builtin_amdgcn_wmma_f32_16x16x4_f32(false, af, false, b0, (short)0, acc[0], false, false);
      acc[1] = __builtin_amdgcn_wmma_f32_16x16x4_f32(false, af, false, b1, (short)0, acc[1], false, false);
      acc[2] = __builtin_amdgcn_wmma_f32_16x16x4_f32(false, af, false, b2, (short)0, acc[2], false, false);
      acc[3] = __builtin_amdgcn_wmma_f32_16x16x4_f32(false, af, false, b3, (short)0, acc[3], false, false);
    }
  }

  // ---- epilogue: bias + ReLU, store h (C layout: VGPR r -> M = 8*hf + r) ----
  const int mrow = row0 + wm * 16 + hf * 8;
#pragma unroll
  for (int nf = 0; nf < 4; ++nf) {
    const int ng = col0 + wn * 64 + nf * 16 + ml;
    const float bs = bias1[ng];
#pragma unroll
    for (int r = 0; r < 8; ++r) {
      const float x = acc[nf][r] + bs;
      hout[(size_t)(mrow + r) * H_DIM + ng] = x > 0.f ? x : 0.f;
    }
  }
}

// ---------------------------------------------------------------------------
// Kernel 2: deterministic batch-norm stats -> fused scale/shift per feature.
// ---------------------------------------------------------------------------
__global__ __launch_bounds__(256) void bn_stats_kernel(
    const float* __restrict__ hbuf, const float* __restrict__ gamma,
    const float* __restrict__ beta, float* __restrict__ scale,
    float* __restrict__ shift)
{
  __shared__ float ssum[4][64];
  __shared__ float ssq[4][64];
  const int cl = threadIdx.x & 63;
  const int strip = threadIdx.x >> 6;
  const int c = blockIdx.x * 64 + cl;
  float s = 0.f, q = 0.f;
  for (int b = strip; b < B_ROWS; b += 4) {
    const float x = hbuf[(size_t)b * H_DIM + c];
    s += x; q += x * x;
  }
  ssum[strip][cl] = s; ssq[strip][cl] = q;
  __syncthreads();
  if (threadIdx.x < 64) {
    const float S = ssum[0][cl] + ssum[1][cl] + ssum[2][cl] + ssum[3][cl];
    const float Q = ssq[0][cl] + ssq[1][cl] + ssq[2][cl] + ssq[3][cl];
    const float mean = S * (1.0f / (float)B_ROWS);
    const float var  = Q * (1.0f / (float)B_ROWS) - mean * mean;
    const float sc = gamma[c] * rsqrtf(var + 1e-5f);
    scale[c] = sc;
    shift[c] = beta[c] - mean * sc;
  }
}

// ---------------------------------------------------------------------------
// Kernel 3: out[4096,21] = (h*scale+shift) @ W2 + b2  (tiny, memory-bound)
// ---------------------------------------------------------------------------
__global__ __launch_bounds__(256) void out_gemm_kernel(
    const float* __restrict__ hbuf, const float* __restrict__ scale,
    const float* __restrict__ shift, const float* __restrict__ W2,
    const float* __restrict__ b2, float* __restrict__ out)
{
  const int idx = blockIdx.x * 256 + threadIdx.x;
  if (idx >= B_ROWS * OUT_D) return;
  const int row = idx / OUT_D, col = idx % OUT_D;
  const float* hr = hbuf + (size_t)row * H_DIM;
  float acc = b2[col];
#pragma unroll 4
  for (int k = 0; k < H_DIM; ++k)
    acc = fmaf(hr[k] * scale[k] + shift[k], W2[k * OUT_D + col], acc);
  out[idx] = acc;
}

// ---------------------------------------------------------------------------
extern "C" void kernel_launch(void* const* d_in, const int* in_sizes, int n_in,
                              void* d_out, int out_size, void* d_ws, size_t ws_size,
                              hipStream_t stream) {
  (void)in_sizes; (void)n_in; (void)out_size; (void)ws_size;
  const float* visual = (const float*)d_in[0];
  const float* audio  = (const float*)d_in[1];
  const float* W1     = (const float*)d_in[2];
  const float* b1     = (const float*)d_in[3];
  const float* gamma  = (const float*)d_in[4];
  const float* beta   = (const float*)d_in[5];
  const float* W2     = (const float*)d_in[6];
  const float* b2     = (const float*)d_in[7];
  float* out   = (float*)d_out;
  float* hbuf  = (float*)d_ws;                 // 4096*1024 floats = 16 MB
  float* scale = hbuf + (size_t)B_ROWS * H_DIM;
  float* shift = scale + H_DIM;

  dim3 grid1(B_ROWS / BM, H_DIM / BN);         // 128 x 8; x walks batch tiles
  fused_bilinear_kernel<<<grid1, 128, 0, stream>>>(visual, audio, W1, b1, hbuf);

  bn_stats_kernel<<<H_DIM / 64, 256, 0, stream>>>(hbuf, gamma, beta, scale, shift);

  const int n_out = B_ROWS * OUT_D;
  out_gemm_kernel<<<(n_out + 255) / 256, 256, 0, stream>>>(hbuf, scale, shift, W2, b2, out);
}